// FlattenLoss_62929860821309
// MI455X (gfx1250) — compile-verified
//
#include <hip/hip_runtime.h>

typedef __attribute__((ext_vector_type(2))) float v2f;
typedef __attribute__((ext_vector_type(8))) float v8f;

#define EPS_F 1e-6f
#define REDUCE_BLOCKS 2048   // multiple of 64 -> no tail in the WMMA reduction
#define TPB 256

__global__ __launch_bounds__(TPB) void flatten_partial_kernel(
    const float* __restrict__ vertices,
    const int*   __restrict__ v0s,
    const int*   __restrict__ v1s,
    const int*   __restrict__ v2s,
    const int*   __restrict__ v3s,
    int nEdges,
    float* __restrict__ partials)
{
    float acc = 0.0f;
    const int stride = gridDim.x * blockDim.x;
    for (int e = blockIdx.x * blockDim.x + threadIdx.x; e < nEdges; e += stride) {
        const int i0 = v0s[e];
        const int i1 = v1s[e];
        const int i2 = v2s[e];
        const int i3 = v3s[e];

        const float* p0 = vertices + 3ll * i0;
        const float* p1 = vertices + 3ll * i1;
        const float* p2 = vertices + 3ll * i2;
        const float* p3 = vertices + 3ll * i3;

        const float p0x = p0[0], p0y = p0[1], p0z = p0[2];
        const float p1x = p1[0], p1y = p1[1], p1z = p1[2];
        const float p2x = p2[0], p2y = p2[1], p2z = p2[2];
        const float p3x = p3[0], p3y = p3[1], p3z = p3[2];

        const float a1x = p1x - p0x, a1y = p1y - p0y, a1z = p1z - p0z;
        const float b1x = p2x - p0x, b1y = p2y - p0y, b1z = p2z - p0z;
        const float b2x = p3x - p0x, b2y = p3y - p0y, b2z = p3z - p0z;

        const float a1l2 = a1x*a1x + a1y*a1y + a1z*a1z;
        const float b1l2 = b1x*b1x + b1y*b1y + b1z*b1z;
        const float b2l2 = b2x*b2x + b2y*b2y + b2z*b2z;

        const float a1l1 = sqrtf(a1l2 + EPS_F);
        const float b1l1 = sqrtf(b1l2 + EPS_F);
        const float b2l1 = sqrtf(b2l2 + EPS_F);

        const float ab1 = a1x*b1x + a1y*b1y + a1z*b1z;
        const float ab2 = a1x*b2x + a1y*b2y + a1z*b2z;

        const float cos1 = ab1 / (a1l1 * b1l1 + EPS_F);
        const float sin1 = sqrtf(1.0f - cos1*cos1 + EPS_F);
        const float r1   = ab1 / (a1l2 + EPS_F);
        const float cb1x = b1x - a1x * r1;
        const float cb1y = b1y - a1y * r1;
        const float cb1z = b1z - a1z * r1;
        const float cb1l1 = b1l1 * sin1;

        const float cos2 = ab2 / (a1l1 * b2l1 + EPS_F);
        const float sin2 = sqrtf(1.0f - cos2*cos2 + EPS_F);
        const float r2   = ab2 / (a1l2 + EPS_F);
        const float cb2x = b2x - a1x * r2;
        const float cb2y = b2y - a1y * r2;
        const float cb2z = b2z - a1z * r2;
        const float cb2l1 = b2l1 * sin2;

        float cosd = (cb1x*cb2x + cb1y*cb2y + cb1z*cb2z) / (cb1l1 * cb2l1 + EPS_F);
        // THRESHOLD_DEG = 0 -> thr = cos(0) = 1.0
        if (cosd > 1.0f) cosd = -1.0f;
        const float t = cosd + 1.0f;
        acc += t * t;
    }

    // wave32 tree reduction
    #pragma unroll
    for (int off = 16; off > 0; off >>= 1)
        acc += __shfl_xor(acc, off, 32);

    __shared__ float wsum[TPB / 32];
    const int lane = threadIdx.x & 31;
    const int wid  = threadIdx.x >> 5;
    if (lane == 0) wsum[wid] = acc;
    __syncthreads();
    if (threadIdx.x == 0) {
        float s = 0.0f;
        #pragma unroll
        for (int i = 0; i < TPB / 32; ++i) s += wsum[i];
        partials[blockIdx.x] = s;
    }
}

// Final reduction of `n` partials (n multiple of 64) on one wave32 using
// V_WMMA_F32_16X16X4_F32 with B = ones:  D = A x 1 + C  ==> per-row f32 sums.
__global__ __launch_bounds__(32) void wmma_reduce_kernel(
    const float* __restrict__ partials, int n, float* __restrict__ out)
{
    const int lane = threadIdx.x;

    v8f acc = {};          // C/D accumulator, starts at zero
    v2f ones;
    ones.x = 1.0f; ones.y = 1.0f;

    for (int base = 0; base < n; base += 64) {
        v2f a;
        a.x = partials[base + 2 * lane + 0];
        a.y = partials[base + 2 * lane + 1];
        // 8 args: (neg_a, A, neg_b, B, c_mod, C, reuse_a, reuse_b)
        acc = __builtin_amdgcn_wmma_f32_16x16x4_f32(
            false, a, false, ones, (short)0, acc, false, false);
    }

    // D[m][n] identical across columns. Lane 0 holds M=0..7 of column 0 in its
    // 8 VGPRs; lane 16 holds M=8..15 of column 0. Sum per-lane, then combine.
    float s = acc[0] + acc[1] + acc[2] + acc[3] + acc[4] + acc[5] + acc[6] + acc[7];
    const float tot = __shfl(s, 0, 32) + __shfl(s, 16, 32);
    if (lane == 0) out[0] = tot;
}

extern "C" void kernel_launch(void* const* d_in, const int* in_sizes, int n_in,
                              void* d_out, int out_size, void* d_ws, size_t ws_size,
                              hipStream_t stream)
{
    const float* vertices = (const float*)d_in[0];
    const int*   v0s      = (const int*)d_in[1];
    const int*   v1s      = (const int*)d_in[2];
    const int*   v2s      = (const int*)d_in[3];
    const int*   v3s      = (const int*)d_in[4];
    const int    nEdges   = in_sizes[1];

    float* partials = (float*)d_ws;   // REDUCE_BLOCKS floats = 8 KB
    float* out      = (float*)d_out;  // single f32 loss

    flatten_partial_kernel<<<REDUCE_BLOCKS, TPB, 0, stream>>>(
        vertices, v0s, v1s, v2s, v3s, nEdges, partials);

    wmma_reduce_kernel<<<1, 32, 0, stream>>>(partials, REDUCE_BLOCKS, out);
}